// Attention_45526653337789
// MI455X (gfx1250) — compile-verified
//
#include <hip/hip_runtime.h>
#include <math.h>

// Bahdanau additive attention for MI455X (gfx1250), FP32 path with
// V_WMMA_F32_16X16X4_F32 for all GEMMs + native V_TANH_F32 for scores.
//
// Shapes: B=4, T=32, S=1024, H=512, 2H=1024, 3H=1536.

typedef float v2f __attribute__((ext_vector_type(2)));
typedef float v8f __attribute__((ext_vector_type(8)));

#define HH   512
#define BB   4
#define TT   32
#define SS   1024
#define D2H  1024
#define D3H  1536

static __device__ __forceinline__ float fast_tanh(float x)
{
#if __has_builtin(__builtin_amdgcn_tanhf)
    return __builtin_amdgcn_tanhf(x);          // v_tanh_f32 (gfx1250 TRANS)
#else
    // tanh(x) = 1 - 2/(exp(2x)+1); exact in both saturation limits:
    // x >> 0: exp->inf -> 1;  x << 0: exp->0 -> -1.
    float e = __expf(2.0f * x);                // v_exp_f32
    return 1.0f - 2.0f / (e + 1.0f);           // v_rcp_f32
#endif
}

// ---------------------------------------------------------------------------
// Generic batched FP32 WMMA GEMM:  C[M,N] = A[M,K] * B[K,N] (+ bias[N])
// Row-major A/B/C. One wave computes a 16(M) x 64(N) strip: 4 accumulators,
// the A fragment is loaded once per k-step and reused for 4 WMMAs.
// Block = 128 threads = 4 waves side by side along N (256 columns / block).
//
// Per-lane layouts (ISA 7.12.2, f32 16x16x4):
//   A (16x4):  lanes 0-15: M=lane, VGPR0=K0/VGPR1=K1; lanes 16-31: M=lane-16,
//              VGPR0=K2/VGPR1=K3.
//   B (4x16):  lanes 0-15: K=0(/1), N=lane; lanes 16-31: K=2(/3), N=lane-16.
//   C (16x16): VGPR i: lanes 0-15 -> M=i, N=lane; lanes 16-31 -> M=i+8.
// ---------------------------------------------------------------------------
__global__ void wmma_gemm_f32(const float* __restrict__ A,
                              const float* __restrict__ Bm,
                              const float* __restrict__ bias,
                              float* __restrict__ C,
                              int M, int N, int K,
                              int lda, int ldb, int ldc,
                              long strideA, long strideB, long strideC)
{
    const int lane = threadIdx.x & 31;
    const int wave = threadIdx.x >> 5;
    const int m0 = blockIdx.x * 16;
    const int n0 = (blockIdx.y * 4 + wave) * 64;
    if (m0 >= M || n0 >= N) return;   // wave-uniform: EXEC stays all-ones

    const float* Ab = A  + (long)blockIdx.z * strideA;
    const float* Bb = Bm + (long)blockIdx.z * strideB;
    float*       Cb = C  + (long)blockIdx.z * strideC;

    const int half = lane >> 4;        // 0: lanes 0-15, 1: lanes 16-31
    const int sub  = lane & 15;
    const int hk   = half << 1;        // K offset 0 or 2 for this half-wave

    const float* arow = Ab + (long)(m0 + sub) * lda + hk;   // A[m][k+hk]
    const float* bptr = Bb + (long)hk * ldb + (n0 + sub);   // B[k+hk][n0+sub]

    v8f acc0 = {}, acc1 = {}, acc2 = {}, acc3 = {};
    #pragma unroll 2
    for (int k = 0; k < K; k += 4) {
        v2f a;
        a.x = arow[k];
        a.y = arow[k + 1];
        const float* bk0 = bptr + (long)k * ldb;
        const float* bk1 = bk0 + ldb;
        v2f b0, b1, b2, b3;
        b0.x = bk0[ 0]; b0.y = bk1[ 0];
        b1.x = bk0[16]; b1.y = bk1[16];
        b2.x = bk0[32]; b2.y = bk1[32];
        b3.x = bk0[48]; b3.y = bk1[48];
        // (neg_a, A, neg_b, B, c_mod, C, reuse_a, reuse_b)
        acc0 = __builtin_amdgcn_wmma_f32_16x16x4_f32(false, a, false, b0,
                                                     (short)0, acc0, false, false);
        acc1 = __builtin_amdgcn_wmma_f32_16x16x4_f32(false, a, false, b1,
                                                     (short)0, acc1, false, false);
        acc2 = __builtin_amdgcn_wmma_f32_16x16x4_f32(false, a, false, b2,
                                                     (short)0, acc2, false, false);
        acc3 = __builtin_amdgcn_wmma_f32_16x16x4_f32(false, a, false, b3,
                                                     (short)0, acc3, false, false);
    }

    const int crow = m0 + half * 8;
    float* crow0 = Cb + (long)crow * ldc + n0 + sub;
    #pragma unroll
    for (int j = 0; j < 4; ++j) {
        const int ccol = n0 + sub + j * 16;
        const float bv = bias ? bias[ccol] : 0.0f;
        const v8f acc = (j == 0) ? acc0 : (j == 1) ? acc1 : (j == 2) ? acc2 : acc3;
        #pragma unroll
        for (int i = 0; i < 8; ++i)
            crow0[(long)i * ldc + j * 16] = acc[i] + bv;
    }
}

// ---------------------------------------------------------------------------
// align[b,t,s] = sum_d v[d] * tanh(qf[b,t,d] + sf[b,s,d]), masked to -inf.
// One block per (b, t, 64-s chunk); qf row and v staged in LDS (reused 64x);
// sf rows streamed as per-lane float4 (b128, coalesced; hot in L2, 16.8 MB).
// 8 waves x 8 s each.
// ---------------------------------------------------------------------------
__global__ void align_kernel(const float* __restrict__ qf,
                             const float* __restrict__ sf,
                             const float* __restrict__ v,
                             const int* __restrict__ mask,
                             float* __restrict__ align)
{
    __shared__ float s_qf[D2H];
    __shared__ float s_v[D2H];
    const int b = blockIdx.z, t = blockIdx.y;
    const int tid = threadIdx.x;

    const float* qrow = qf + (long)(b * TT + t) * D2H;
    for (int i = tid; i < D2H; i += 256) { s_qf[i] = qrow[i]; s_v[i] = v[i]; }
    __syncthreads();

    const int wave = tid >> 5, lane = tid & 31;
    for (int j = 0; j < 8; ++j) {
        const int s = blockIdx.x * 64 + wave * 8 + j;
        const float* srow = sf + ((long)b * SS + s) * D2H;
        float acc = 0.0f;
        #pragma unroll
        for (int d0 = lane * 4; d0 < D2H; d0 += 128) {
            const float4 sv = *(const float4*)(srow + d0);
            const float4 qv = *(const float4*)(s_qf + d0);
            const float4 vv = *(const float4*)(s_v + d0);
            acc += vv.x * fast_tanh(qv.x + sv.x);
            acc += vv.y * fast_tanh(qv.y + sv.y);
            acc += vv.z * fast_tanh(qv.z + sv.z);
            acc += vv.w * fast_tanh(qv.w + sv.w);
        }
        #pragma unroll
        for (int off = 16; off > 0; off >>= 1)
            acc += __shfl_xor(acc, off, 32);
        if (lane == 0) {
            float out = mask[b * SS + s] ? acc : -__builtin_inff();
            align[(long)(b * TT + t) * SS + s] = out;
        }
    }
}

// ---------------------------------------------------------------------------
// Row softmax over S=1024 (in-place on attn) + scatter of transposed
// attentions attT[b,s,t]. One 256-thread block per (b,t) row, 4 elems/thread.
// ---------------------------------------------------------------------------
__global__ void softmax_kernel(float* __restrict__ attn,
                               float* __restrict__ attT)
{
    __shared__ float red_m[8];
    __shared__ float red_s[8];
    const int row = blockIdx.x;              // b*T + t
    const int b = row / TT, t = row % TT;
    const int tid = threadIdx.x, lane = tid & 31, wave = tid >> 5;
    float* arow = attn + (long)row * SS;

    float vals[4];
    float m = -__builtin_inff();
    #pragma unroll
    for (int i = 0; i < 4; ++i) {
        vals[i] = arow[tid + i * 256];
        m = fmaxf(m, vals[i]);
    }
    #pragma unroll
    for (int off = 16; off > 0; off >>= 1)
        m = fmaxf(m, __shfl_xor(m, off, 32));
    if (lane == 0) red_m[wave] = m;
    __syncthreads();
    m = red_m[0];
    #pragma unroll
    for (int i = 1; i < 8; ++i) m = fmaxf(m, red_m[i]);

    float sum = 0.0f;
    #pragma unroll
    for (int i = 0; i < 4; ++i) {
        vals[i] = __expf(vals[i] - m);
        sum += vals[i];
    }
    #pragma unroll
    for (int off = 16; off > 0; off >>= 1)
        sum += __shfl_xor(sum, off, 32);
    if (lane == 0) red_s[wave] = sum;
    __syncthreads();
    float total = red_s[0];
    #pragma unroll
    for (int i = 1; i < 8; ++i) total += red_s[i];
    const float inv = 1.0f / total;

    #pragma unroll
    for (int i = 0; i < 4; ++i) {
        const int s = tid + i * 256;
        const float p = vals[i] * inv;
        arow[s] = p;                                    // for context GEMM
        attT[((long)b * SS + s) * TT + t] = p;          // output 3 (B,S,T)
    }
}

// ---------------------------------------------------------------------------
// concat[(b*T+t), 0:3H] = { context[b,t,0:2H], query[b,t,0:H] }
// ---------------------------------------------------------------------------
__global__ void concat_kernel(const float* __restrict__ context,
                              const float* __restrict__ query,
                              float* __restrict__ cc)
{
    const long idx = (long)blockIdx.x * 256 + threadIdx.x;  // B*T*3H total
    const int  j   = (int)(idx % D3H);
    const long row = idx / D3H;                              // b*T + t
    cc[idx] = (j < D2H) ? context[row * D2H + j]
                        : query[row * HH + (j - D2H)];
}

// ---------------------------------------------------------------------------
extern "C" void kernel_launch(void* const* d_in, const int* in_sizes, int n_in,
                              void* d_out, int out_size, void* d_ws, size_t ws_size,
                              hipStream_t stream)
{
    const float* query  = (const float*)d_in[0];   // (B,T,H)
    const float* states = (const float*)d_in[1];   // (B,S,2H)
    const int*   mask   = (const int*)  d_in[2];   // (B,S)
    const float* Wq     = (const float*)d_in[3];   // (H,2H)
    const float* bq     = (const float*)d_in[4];   // (2H)
    const float* Ws     = (const float*)d_in[5];   // (2H,2H)
    const float* v      = (const float*)d_in[6];   // (2H)
    const float* Wc     = (const float*)d_in[7];   // (3H,H)
    const float* bc     = (const float*)d_in[8];   // (H)

    float* out     = (float*)d_out;
    float* context = out;                                        // B*T*2H
    float* hidden  = out + (long)BB * TT * D2H;                  // B*T*H
    float* attT    = hidden + (long)BB * TT * HH;                // B*S*T

    float* ws   = (float*)d_ws;
    float* qf   = ws;                                            // 131072
    float* sf   = qf   + (long)BB * TT * D2H;                    // 4194304
    float* attn = sf   + (long)BB * SS * D2H;                    // 131072
    float* cc   = attn + (long)BB * TT * SS;                     // 196608

    // 1) qf = query @ Wq + bq            M=128  N=1024 K=512
    wmma_gemm_f32<<<dim3(8, 4, 1), 128, 0, stream>>>(
        query, Wq, bq, qf, 128, 1024, 512, 512, 1024, 1024, 0, 0, 0);

    // 2) sf = states @ Ws                M=4096 N=1024 K=1024
    wmma_gemm_f32<<<dim3(256, 4, 1), 128, 0, stream>>>(
        states, Ws, nullptr, sf, 4096, 1024, 1024, 1024, 1024, 1024, 0, 0, 0);

    // 3) additive-attention scores (tanh reduction) + mask
    align_kernel<<<dim3(SS / 64, TT, BB), 256, 0, stream>>>(
        qf, sf, v, mask, attn);

    // 4) softmax over S (in-place) + transposed attention output
    softmax_kernel<<<dim3(BB * TT), 256, 0, stream>>>(attn, attT);

    // 5) context = attn @ states  (batched over B)  M=32 N=1024 K=1024
    wmma_gemm_f32<<<dim3(2, 4, BB), 128, 0, stream>>>(
        attn, states, nullptr, context, 32, 1024, 1024, 1024, 1024, 1024,
        (long)TT * SS, (long)SS * D2H, (long)TT * D2H);

    // 6) concat(context, query)
    concat_kernel<<<dim3((BB * TT * D3H) / 256), 256, 0, stream>>>(
        context, query, cc);

    // 7) hidden = concat @ Wc + bc       M=128  N=512  K=1536
    wmma_gemm_f32<<<dim3(8, 2, 1), 128, 0, stream>>>(
        cc, Wc, bc, hidden, 128, 512, 1536, 1536, 512, 512, 0, 0, 0);
}